// PairwiseRankingLoss_73057393705015
// MI455X (gfx1250) — compile-verified
//
#include <hip/hip_runtime.h>

#define NN  1024
#define BLK 256
#define MARGIN 1.0f

typedef __attribute__((ext_vector_type(2))) float v2f;
typedef __attribute__((ext_vector_type(8))) float v8f;

// Accumulate the sum of 64 LDS floats into a 16x16 f32 WMMA accumulator using
// V_WMMA_F32_16X16X4_F32 with an all-ones 4x16 B matrix.  D[m,n] += sum_k A[m,k].
// Must be called by one full wave (EXEC all ones).  Since B is all-ones and we
// later sum the entire accumulator, any bijective placement of the 64 values
// into the 16x4 A tile produces the same grand total (layout-robust).
__device__ inline v8f wmma_accum64(const float* __restrict__ base, int lane, v8f c) {
  const int m  = lane & 15;   // A-matrix row held by this lane
  const int kh = lane >> 4;   // lanes 0-15 hold K={0,1}, lanes 16-31 hold K={2,3}
  v2f a;
  a.x = base[m * 4 + kh * 2 + 0];
  a.y = base[m * 4 + kh * 2 + 1];
  v2f b;
  b.x = 1.0f;
  b.y = 1.0f;
  // 8 args: (neg_a, A, neg_b, B, c_mod, C, reuse_a, reuse_b)
  return __builtin_amdgcn_wmma_f32_16x16x4_f32(false, a, false, b, (short)0, c,
                                               false, false);
}

// Collapse the 16x16 accumulator to a grand total.  Per the C/D layout, lane l
// (l<16) holds C[0..7, l] in its 8 VGPRs and lane l+16 holds C[8..15, l]; an
// 8-element sum + shfl_xor(16) yields sum over all M for column l, i.e. the
// full total (all columns are identical because B was all-ones).
__device__ inline float wmma_total(v8f c) {
  float s = c[0] + c[1] + c[2] + c[3] + c[4] + c[5] + c[6] + c[7];
  s += __shfl_xor(s, 16, 32);
  return s;
}

__global__ __launch_bounds__(BLK)
void pairloss_sample_kernel(const float* __restrict__ pred,
                            const float* __restrict__ targ,
                            float* __restrict__ wsLoss,
                            float* __restrict__ wsOk) {
  __shared__ float4 pt[NN];        // {p, nan_to_num(t), valid, 0}
  __shared__ float  partialL[BLK];
  __shared__ float  partialC[BLK];

  const int b   = blockIdx.x;
  const int tid = threadIdx.x;
  const float* __restrict__ p = pred + (size_t)b * NN;
  const float* __restrict__ t = targ + (size_t)b * NN;

  // Stage sample into LDS; count valid entries per thread.
  float cntAcc = 0.0f;
  for (int j = tid; j < NN; j += BLK) {
    float pv = p[j];
    float tv = t[j];
    bool valid = !(tv != tv);            // !isnan
    float v = valid ? 1.0f : 0.0f;
    pt[j] = make_float4(pv, valid ? tv : 0.0f, v, 0.0f);
    cntAcc += v;
  }
  __syncthreads();

  // Pairwise loop.  j is uniform across the wave -> LDS b128 broadcast loads.
  float acc = 0.0f;
  for (int i = tid; i < NN; i += BLK) {
    float4 qi = pt[i];
    if (qi.z == 0.0f) continue;          // invalid i contributes nothing
    const float pi = qi.x;
    const float ti = qi.y;
#pragma unroll 4
    for (int j = 0; j < NN; ++j) {
      float4 qj = pt[j];
      float d = pi - qj.x;               // pred_i - pred_j
      float l = (ti > qj.y) ? fmaxf(MARGIN - d, 0.0f)
                            : fmaxf(MARGIN + d, 0.0f);
      float w = (j == i) ? 0.0f : qj.z;  // off-diagonal valid-pair mask
      acc = fmaf(w, l, acc);
    }
  }

  partialL[tid] = acc;
  partialC[tid] = cntAcc;
  __syncthreads();

  // Wave 0 (uniform branch -> EXEC all ones) reduces 256 partials with WMMA.
  if (tid < 32) {
    v8f cL = {};
    v8f cC = {};
#pragma unroll
    for (int tile = 0; tile < BLK / 64; ++tile) {
      cL = wmma_accum64(partialL + tile * 64, tid, cL);
      cC = wmma_accum64(partialC + tile * 64, tid, cC);
    }
    float totalL = wmma_total(cL);
    float nv     = wmma_total(cC);       // exact small integer in f32
    if (tid == 0) {
      float cnt = nv * (nv - 1.0f);      // number of valid ordered pairs
      float sl  = totalL / fmaxf(cnt, 1.0f);
      bool  ok  = nv > 1.0f;             // samples with >= 2 valid items
      wsLoss[b] = ok ? sl : 0.0f;
      wsOk[b]   = ok ? 1.0f : 0.0f;
    }
  }
}

__global__ __launch_bounds__(32)
void pairloss_final_kernel(const float* __restrict__ wsLoss,
                           const float* __restrict__ wsOk,
                           float* __restrict__ out, int B) {
  __shared__ float smL[64];
  __shared__ float smO[64];
  const int lane = threadIdx.x;

  v8f cL = {};
  v8f cO = {};
  for (int b0 = 0; b0 < B; b0 += 64) {   // B==64: one 16x4 A tile per array
    for (int k = lane; k < 64; k += 32) {
      int b = b0 + k;
      smL[k] = (b < B) ? wsLoss[b] : 0.0f;
      smO[k] = (b < B) ? wsOk[b]   : 0.0f;
    }
    __syncthreads();
    cL = wmma_accum64(smL, lane, cL);
    cO = wmma_accum64(smO, lane, cO);
    __syncthreads();
  }
  float sumL  = wmma_total(cL);
  float numOk = wmma_total(cO);
  if (lane == 0) {
    // masked loss sum is 0 when numOk==0, so /max(numOk,1) matches jnp.where
    out[0] = sumL / fmaxf(numOk, 1.0f);
  }
}

extern "C" void kernel_launch(void* const* d_in, const int* in_sizes, int n_in,
                              void* d_out, int out_size, void* d_ws, size_t ws_size,
                              hipStream_t stream) {
  const float* pred = (const float*)d_in[0];
  const float* targ = (const float*)d_in[1];
  const int B = in_sizes[0] / NN;        // 64 for the reference shapes

  float* wsLoss = (float*)d_ws;          // [B]
  float* wsOk   = wsLoss + B;            // [B]

  pairloss_sample_kernel<<<B, BLK, 0, stream>>>(pred, targ, wsLoss, wsOk);
  pairloss_final_kernel<<<1, 32, 0, stream>>>(wsLoss, wsOk, (float*)d_out, B);
}